// Net_corr_50414326121019
// MI455X (gfx1250) — compile-verified
//
#include <hip/hip_runtime.h>
#include <math.h>

// Row-L2-normalize an 8192x8192 fp32 matrix:
//   out[i,:] = corr[i,:] * rsqrt(sum_j corr[i,j]^2)
//
// Bandwidth-bound: 512 MB min HBM traffic -> ~22 us floor @ 23.3 TB/s.
// Strategy: 1 workgroup per row (32 KB), stage the row once into LDS with
// CDNA5 async global->LDS copies (ASYNCcnt), wave32 butterfly reduction,
// then scale + non-temporal streaming stores. Input bytes are read from HBM
// exactly once; output written exactly once.

typedef float        v4f  __attribute__((ext_vector_type(4)));
typedef int          v4i  __attribute__((ext_vector_type(4)));
typedef unsigned int u32;
typedef unsigned long long u64;

#define DIM     8192
#define BLOCK   256               // 8 waves (wave32)
#define WAVES   (BLOCK / 32)
#define CHUNK   (BLOCK * 4)       // 1024 floats per block-wide b128 pass
#define CHUNKS  (DIM / CHUNK)     // 8

#if __has_builtin(__builtin_amdgcn_global_load_async_to_lds_b128)
#define STAGE_ASYNC 1
#else
#define STAGE_ASYNC 0             // host pass / older toolchains
#endif

// Builtin signature (from clang diagnostic): (v4i AS1*, v4i AS3*, imm, imm)
typedef __attribute__((address_space(1))) v4i* gas1_v4i;
typedef __attribute__((address_space(3))) v4i* las3_v4i;

__global__ __launch_bounds__(BLOCK) void rownorm_scale_kernel(
    const float* __restrict__ corr, float* __restrict__ out) {
  const int tid = threadIdx.x;
  const int row = blockIdx.x;
  const float* __restrict__ rp = corr + (size_t)row * DIM;
  float* __restrict__ op = out + (size_t)row * DIM;

  __shared__ float wpart[WAVES];

#if STAGE_ASYNC
  __shared__ v4f lrow[DIM / 4];   // 32 KB: the whole row staged in LDS
  // Per-lane 16B async global->LDS copies; 8 block-wide passes = 32 KB.
  // Tracked on ASYNCcnt; data bypasses VGPRs on the way to LDS.
  #pragma unroll
  for (int c = 0; c < CHUNKS; ++c) {
    const int idx = c * CHUNK + tid * 4;
    gas1_v4i g = (gas1_v4i)(u64)(rp + idx);
    las3_v4i l = (las3_v4i)(u32)(u64)&lrow[idx >> 2]; // low 32 bits = LDS offset
    __builtin_amdgcn_global_load_async_to_lds_b128(g, l, 0, 0);
  }
  #if __has_builtin(__builtin_amdgcn_s_wait_asynccnt)
  __builtin_amdgcn_s_wait_asynccnt(0);
  #else
  asm volatile("s_wait_asynccnt 0" ::: "memory");
  #endif
  __syncthreads();
#endif

  // ---- pass 1: sum of squares (row pulled into registers once) ----
  v4f v[CHUNKS];
  float s = 0.0f;
  #pragma unroll
  for (int c = 0; c < CHUNKS; ++c) {
    const int idx = c * CHUNK + tid * 4;
#if STAGE_ASYNC
    v[c] = lrow[idx >> 2];                                     // ds_load_b128
#else
    v[c] = __builtin_nontemporal_load((const v4f*)(rp + idx)); // NT b128
#endif
    s = fmaf(v[c].x, v[c].x, s);
    s = fmaf(v[c].y, v[c].y, s);
    s = fmaf(v[c].z, v[c].z, s);
    s = fmaf(v[c].w, v[c].w, s);
  }

  // wave32 butterfly reduction
  #pragma unroll
  for (int o = 16; o > 0; o >>= 1) s += __shfl_xor(s, o, 32);
  if ((tid & 31) == 0) wpart[tid >> 5] = s;
  __syncthreads();

  // every thread sums the 8 wave partials in the same order (deterministic)
  float tot = 0.0f;
  #pragma unroll
  for (int w = 0; w < WAVES; ++w) tot += wpart[w];
  const float inv = 1.0f / sqrtf(tot);

  // ---- pass 2: scale + non-temporal streaming stores ----
  #pragma unroll
  for (int c = 0; c < CHUNKS; ++c) {
    const int idx = c * CHUNK + tid * 4;
    v4f r;
    r.x = v[c].x * inv;
    r.y = v[c].y * inv;
    r.z = v[c].z * inv;
    r.w = v[c].w * inv;
    __builtin_nontemporal_store(r, (v4f*)(op + idx));
  }
}

extern "C" void kernel_launch(void* const* d_in, const int* in_sizes, int n_in,
                              void* d_out, int out_size, void* d_ws, size_t ws_size,
                              hipStream_t stream) {
  (void)in_sizes; (void)n_in; (void)d_ws; (void)ws_size; (void)out_size;
  const float* corr = (const float*)d_in[0];
  float* out = (float*)d_out;
  rownorm_scale_kernel<<<DIM, BLOCK, 0, stream>>>(corr, out);
}